// UnifiedSimilarity_7035156431350
// MI455X (gfx1250) — compile-verified
//
#include <hip/hip_runtime.h>

#define BLOCK  256
#define WAVES  (BLOCK / 32)
#define WCHUNK 128   // elements per wave per chunk: lane*4 dwords via async B128

// ---- ordered-uint encoding so float max can use hardware u32 atomicMax ----
static __device__ __forceinline__ unsigned encf(float f) {
    unsigned u = __float_as_uint(f);
    return (u & 0x80000000u) ? ~u : (u | 0x80000000u);
}
static __device__ __forceinline__ float decf(unsigned u) {
    unsigned v = (u & 0x80000000u) ? (u & 0x7fffffffu) : ~u;
    return __uint_as_float(v);
}

// packed per-row table: pk[r].x = ordered rowmax bits, pk[r].y = denom (f32 bits)
// -> one B64 gather fetches both in the normalize pass.

// ---------------------------------------------------------------------------
// Kernel 1: zero/neutral init (B64 stores for the packed table, B128 for edge_w)
// ---------------------------------------------------------------------------
__global__ void k_init(uint2* __restrict__ pk,
                       float* __restrict__ degree,
                       float4* __restrict__ edge_w4,
                       float* __restrict__ edge_w,
                       int n, int n_und) {
    int i = blockIdx.x * blockDim.x + threadIdx.x;
    if (i < n) {
        pk[i]     = make_uint2(0u, 0u);   // rowmax = "-inf", denom = 0.0f
        degree[i] = 0.0f;
    }
    int nq = n_und >> 2;
    if (i < nq) {
        edge_w4[i] = make_float4(0.f, 0.f, 0.f, 0.f);
    }
    if (i == 0) {
        for (int e = nq << 2; e < n_und; ++e) edge_w[e] = 0.0f;
    }
}

// ---------------------------------------------------------------------------
// Kernels 2+3 (templated): segmented reduction over sorted row_index with a
// double-buffered ASYNC global->LDS pipeline (gfx1250 ASYNCcnt path).
//   OP == 0 : pk[r].x = max(theta)            (ordered-u32 atomicMax)
//   OP == 1 : pk[r].y += exp(theta - max)     (global_atomic_add_f32)
// Each wave stages its own 128-element slice of theta/row into LDS with two
// GLOBAL_LOAD_ASYNC_TO_LDS_B128 ops, prefetching chunk t+1 while reducing
// chunk t. Only the tail lane of each sorted run issues a global atomic.
// ---------------------------------------------------------------------------
template <int OP>
__global__ void k_seg_reduce(const float* __restrict__ theta,
                             const int* __restrict__ row,
                             uint2* __restrict__ pk,
                             int nnz) {
    __shared__ float th_buf[2][WAVES * WCHUNK];
    __shared__ int   rw_buf[2][WAVES * WCHUNK];

    const int lane   = threadIdx.x & 31;
    const int wid    = threadIdx.x >> 5;
    const int gwave  = blockIdx.x * WAVES + wid;
    const int nwaves = gridDim.x * WAVES;
    const int stride = nwaves * WCHUNK;   // elements advanced per wave-iteration

    float* tb0 = &th_buf[0][wid * WCHUNK];
    float* tb1 = &th_buf[1][wid * WCHUNK];
    int*   rb0 = &rw_buf[0][wid * WCHUNK];
    int*   rb1 = &rw_buf[1][wid * WCHUNK];

    const int start0 = gwave * WCHUNK;
    if (start0 >= nnz) return;            // wave-uniform early exit

    // issue async copy of one full chunk (start+WCHUNK <= nnz) into buffer b
    auto issue = [&](int start, int b) {
        unsigned long long gth =
            (unsigned long long)(const void*)(theta + start + lane * 4);
        unsigned long long grw =
            (unsigned long long)(const void*)(row + start + lane * 4);
        unsigned lth = (unsigned)(unsigned long long)(const void*)
                           ((b ? tb1 : tb0) + lane * 4);
        unsigned lrw = (unsigned)(unsigned long long)(const void*)
                           ((b ? rb1 : rb0) + lane * 4);
        asm volatile("global_load_async_to_lds_b128 %0, %1, off"
                     :: "v"(lth), "v"(gth) : "memory");
        asm volatile("global_load_async_to_lds_b128 %0, %1, off"
                     :: "v"(lrw), "v"(grw) : "memory");
    };

    const bool full0 = (start0 + WCHUNK) <= nnz;
    if (full0) issue(start0, 0);          // prologue: stage chunk 0

    int buf = 0;
    for (int start = start0; start < nnz; start += stride, buf ^= 1) {
        const int  nstart   = start + stride;
        const bool cur_full = (start + WCHUNK) <= nnz;
        const bool nxt_full = (nstart < nnz) && ((nstart + WCHUNK) <= nnz);

        if (nxt_full) {
            issue(nstart, buf ^ 1);       // prefetch chunk t+1 (other buffer)
            asm volatile("s_wait_asynccnt 0x2" ::: "memory");  // chunk t landed
        } else {
            asm volatile("s_wait_asynccnt 0x0" ::: "memory");
        }

        const float* tb = buf ? tb1 : tb0;
        const int*   rb = buf ? rb1 : rb0;

        #pragma unroll
        for (int s = 0; s < WCHUNK / 32; ++s) {
            const int idx = s * 32 + lane;
            const int e   = start + idx;
            int   r = -1;
            float v = (OP == 0) ? -3.402823466e38f : 0.0f;
            if (e < nnz) {
                float t;
                if (cur_full) { t = tb[idx]; r = rb[idx]; }   // LDS (ds_load)
                else          { t = theta[e]; r = row[e]; }   // guarded tail
                if (OP == 0) {
                    v = t;
                } else {
                    v = expf(t - decf(pk[r].x));              // table L2-hot
                }
            }
            // inclusive wave32 segmented scan over sorted keys
            #pragma unroll
            for (int o = 1; o < 32; o <<= 1) {
                float nv = __shfl_up(v, o);
                int   nr = __shfl_up(r, o);
                if (lane >= o && nr == r) {
                    v = (OP == 0) ? fmaxf(v, nv) : (v + nv);
                }
            }
            const int nxt = __shfl_down(r, 1);
            if (r >= 0 && (lane == 31 || nxt != r)) {
                if (OP == 0) atomicMax(&pk[r].x, encf(v));
                else         atomicAdd((float*)&pk[r].y, v);
            }
        }
    }
}

// ---------------------------------------------------------------------------
// Kernel 4: u = exp(theta - max)/denom, write u_data (B128), scatter 0.5*u.
// 4 edges per thread; per-row (max,denom) fetched with a single B64 gather.
// ---------------------------------------------------------------------------
__global__ void k_u_edgew(const float* __restrict__ theta,
                          const int* __restrict__ row,
                          const int* __restrict__ umap,
                          const uint2* __restrict__ pk,
                          float* __restrict__ u_out,
                          float* __restrict__ edge_w,
                          int nnz) {
    const int q  = blockIdx.x * blockDim.x + threadIdx.x;
    const int nq = nnz >> 2;
    if (q < nq) {
        __builtin_prefetch(theta + (q + 4096) * 4, 0, 0);     // global_prefetch_b8
        float4 t = reinterpret_cast<const float4*>(theta)[q]; // B128
        int4   r = reinterpret_cast<const int4*>(row)[q];
        int4   m = reinterpret_cast<const int4*>(umap)[q];
        float4 u;
        {
            uint2 p = pk[r.x];
            u.x = expf(t.x - decf(p.x)) / fmaxf(__uint_as_float(p.y), 1e-12f);
        }
        {
            uint2 p = pk[r.y];
            u.y = expf(t.y - decf(p.x)) / fmaxf(__uint_as_float(p.y), 1e-12f);
        }
        {
            uint2 p = pk[r.z];
            u.z = expf(t.z - decf(p.x)) / fmaxf(__uint_as_float(p.y), 1e-12f);
        }
        {
            uint2 p = pk[r.w];
            u.w = expf(t.w - decf(p.x)) / fmaxf(__uint_as_float(p.y), 1e-12f);
        }
        reinterpret_cast<float4*>(u_out)[q] = u;              // B128
        atomicAdd(&edge_w[m.x], 0.5f * u.x);                  // global_atomic_add_f32
        atomicAdd(&edge_w[m.y], 0.5f * u.y);
        atomicAdd(&edge_w[m.z], 0.5f * u.z);
        atomicAdd(&edge_w[m.w], 0.5f * u.w);
    }
    if (q == 0) {                                             // scalar tail (<=3)
        for (int e = nq << 2; e < nnz; ++e) {
            uint2 p = pk[row[e]];
            float u = expf(theta[e] - decf(p.x)) /
                      fmaxf(__uint_as_float(p.y), 1e-12f);
            u_out[e] = u;
            atomicAdd(&edge_w[umap[e]], 0.5f * u);
        }
    }
}

// ---------------------------------------------------------------------------
// Kernel 5: weighted degree via hardware f32 atomics; B128 streaming reads,
// 4 edges per thread, L2-resident 400KB target table.
// ---------------------------------------------------------------------------
__global__ void k_degree(const float* __restrict__ edge_w,
                         const int* __restrict__ ei,
                         const int* __restrict__ ej,
                         float* __restrict__ degree,
                         int n_und) {
    const int q  = blockIdx.x * blockDim.x + threadIdx.x;
    const int nq = n_und >> 2;
    if (q < nq) {
        float4 w = reinterpret_cast<const float4*>(edge_w)[q]; // B128
        int4   a = reinterpret_cast<const int4*>(ei)[q];
        int4   b = reinterpret_cast<const int4*>(ej)[q];
        atomicAdd(&degree[a.x], w.x);  atomicAdd(&degree[b.x], w.x);
        atomicAdd(&degree[a.y], w.y);  atomicAdd(&degree[b.y], w.y);
        atomicAdd(&degree[a.z], w.z);  atomicAdd(&degree[b.z], w.z);
        atomicAdd(&degree[a.w], w.w);  atomicAdd(&degree[b.w], w.w);
    }
    if (q == 0) {                                              // scalar tail (<=3)
        for (int e = nq << 2; e < n_und; ++e) {
            float w = edge_w[e];
            atomicAdd(&degree[ei[e]], w);
            atomicAdd(&degree[ej[e]], w);
        }
    }
}

extern "C" void kernel_launch(void* const* d_in, const int* in_sizes, int n_in,
                              void* d_out, int out_size, void* d_ws, size_t ws_size,
                              hipStream_t stream) {
    (void)n_in; (void)ws_size;
    const float* theta = (const float*)d_in[0];
    const int*   row   = (const int*)d_in[1];
    const int*   umap  = (const int*)d_in[2];
    const int*   ei    = (const int*)d_in[3];
    const int*   ej    = (const int*)d_in[4];
    // d_in[5] = num_nodes (device scalar) -- derive N from sizes instead.

    const int nnz   = in_sizes[0];
    const int n_und = in_sizes[3];
    const int n     = out_size - nnz - n_und;   // num_nodes

    // d_out layout: [u_data | edge_w | degree]
    float* u_out  = (float*)d_out;
    float* edge_w = u_out + nnz;
    float* degree = edge_w + n_und;

    // ws layout: interleaved per-row table pk[n] = {rowmax_bits, denom_bits}
    uint2* pk = (uint2*)d_ws;

    const int nq_und      = n_und >> 2;
    const int init_len    = (n > nq_und) ? n : nq_und;
    const int init_blocks = (init_len + BLOCK - 1) / BLOCK;
    k_init<<<init_blocks, BLOCK, 0, stream>>>(pk, degree, (float4*)edge_w,
                                              edge_w, n, n_und);

    // ~6 chunks per wave at 1024 blocks -> real double-buffer pipelining
    int seg_blocks = (nnz + WAVES * WCHUNK - 1) / (WAVES * WCHUNK);
    if (seg_blocks > 1024) seg_blocks = 1024;
    k_seg_reduce<0><<<seg_blocks, BLOCK, 0, stream>>>(theta, row, pk, nnz);
    k_seg_reduce<1><<<seg_blocks, BLOCK, 0, stream>>>(theta, row, pk, nnz);

    const int u_blocks = ((nnz >> 2) + BLOCK) / BLOCK + 1;
    k_u_edgew<<<u_blocks, BLOCK, 0, stream>>>(theta, row, umap, pk,
                                              u_out, edge_w, nnz);

    const int d_blocks = ((n_und >> 2) + BLOCK) / BLOCK + 1;
    k_degree<<<d_blocks, BLOCK, 0, stream>>>(edge_w, ei, ej, degree, n_und);
}